// CRsAE1d_72516227826030
// MI455X (gfx1250) — compile-verified
//
#include <hip/hip_runtime.h>
#include <math.h>

// ---------------------------------------------------------------------------
// CRsAE1d (FISTA convolutional sparse coding) for MI455X / gfx1250.
//
// H@w and H^T@v are expressed as small GEMMs with fixed 16x80 Toeplitz
// A-matrices (per filter) so both ops run on V_WMMA_F32_16X16X4_F32.
// Toeplitz banks are stored row-major [r][u] so each lane's A fragment is one
// ds_load_b64. Pure-copy staging uses GLOBAL_LOAD_ASYNC_TO_LDS_B128.
// L = max_w sum_k |DFT(f_k)(w)|^2 + 1 (exact eigenvalue of circulant Gram).
// Problem is L2-resident (~9 MB state vs 192 MB L2); BW floor ~15us.
// ---------------------------------------------------------------------------

typedef __attribute__((ext_vector_type(2))) float v2f;
typedef __attribute__((ext_vector_type(8))) float v8f;
typedef __attribute__((ext_vector_type(4))) int   v4i;

#define NN 2048          // signal length
#define KF 8             // number of filters
#define KS 64            // filter length
#define NB 64            // batch (columns)
#define T_ITERS 15
#define LAMBDA 0.1f
#define WIN 80           // contraction window per filter (16+64 padded to 80)
#define WINP 84          // padded A row stride (dwords): conflict-free b64 loads
#define LSTRIDE 68       // LDS row stride (floats) for B tile

// workspace layout (float offsets)
#define XELEMS (KF*NN*NB)            // 1,048,576 floats (4 MB)
#define VELEMS (NN*NB)               // 131,072
#define ATELEMS (KF*16*WINP)         // 10,752 (transposed Toeplitz [k][r][u])
#define OFF_X0  ((size_t)0)
#define OFF_X1  ((size_t)XELEMS)
#define OFF_V   ((size_t)(2*XELEMS))
#define OFF_AT  (OFF_V + VELEMS)     // Toeplitz for H@w   : [k][r][u]
#define OFF_AT2 (OFF_AT + ATELEMS)   // Toeplitz for H^T@v : [k][r][u]
#define OFF_L   (OFF_AT2 + ATELEMS)  // 1 float (max eigenvalue, via uint max)

// ---------------------------------------------------------------------------
// CDNA5 async global->LDS staging (ASYNCcnt-tracked, no VGPR round trip).
#if defined(__has_builtin)
#if __has_builtin(__builtin_amdgcn_global_load_async_to_lds_b128)
#define HAS_ASYNC_LDS 1
#endif
#endif
#ifndef HAS_ASYNC_LDS
#define HAS_ASYNC_LDS 0
#endif

typedef __attribute__((address_space(1))) v4i* gvec_p;  // global 128-bit vec ptr
typedef __attribute__((address_space(3))) v4i* svec_p;  // LDS 128-bit vec ptr

__device__ __forceinline__ void stage16(const float* __restrict__ g, float* l) {
#if HAS_ASYNC_LDS
    __builtin_amdgcn_global_load_async_to_lds_b128((gvec_p)g, (svec_p)l, 0, 0);
#else
    *(float4*)l = *(const float4*)g;
#endif
}

__device__ __forceinline__ void stage_wait() {
#if HAS_ASYNC_LDS
#if __has_builtin(__builtin_amdgcn_s_wait_asynccnt)
    __builtin_amdgcn_s_wait_asynccnt(0);
#else
    asm volatile("s_wait_asynccnt 0" ::: "memory");
#endif
#endif
}

// ---------------------------------------------------------------------------
__global__ void k_init(float* __restrict__ x0, float* __restrict__ x1,
                       unsigned int* __restrict__ lval, int n4) {
    int i = blockIdx.x * blockDim.x + threadIdx.x;
    if (i == 0) *lval = 0u;
    float4 z = make_float4(0.f, 0.f, 0.f, 0.f);
    for (int idx = i; idx < n4; idx += gridDim.x * blockDim.x) {
        ((float4*)x0)[idx] = z;
        ((float4*)x1)[idx] = z;
    }
}

// Build the two fixed Toeplitz operand banks, TRANSPOSED: [k][r][u], row
// stride WINP so per-lane A fragments are adjacent (one ds_load_b64).
//   at [k][r][u] = f_k[r + 63 - u]   (H @ w : out[i] = sum_t f[t] w[i-t])
//   at2[k][r][u] = f_k[u - r]        (H^T@v : out[n] = sum_t f[t] v[n+t])
__global__ void k_toep(const float* __restrict__ f,
                       float* __restrict__ at, float* __restrict__ at2) {
    for (int idx = blockIdx.x * blockDim.x + threadIdx.x; idx < ATELEMS;
         idx += gridDim.x * blockDim.x) {
        int u = idx % WINP;
        int rr = idx / WINP;
        int r = rr & 15;
        int k = rr >> 4;
        int t1 = r + 63 - u;
        at[idx]  = (u < WIN && t1 >= 0 && t1 < KS) ? f[k * KS + t1] : 0.f;
        int t2 = u - r;
        at2[idx] = (u < WIN && t2 >= 0 && t2 < KS) ? f[k * KS + t2] : 0.f;
    }
}

// lambda_max(H H^T) = max_w sum_k |fhat_k(w)|^2 (circulant Gram is
// DFT-diagonal). 2048 freqs, exact phase via (w*t) mod N.
__global__ void k_eig(const float* __restrict__ f, unsigned int* __restrict__ lval) {
    int w = blockIdx.x * blockDim.x + threadIdx.x;  // 0..2047
    float sum = 0.f;
    for (int k = 0; k < KF; ++k) {
        float re = 0.f, im = 0.f;
        for (int t = 0; t < KS; ++t) {
            int ph = (w * t) & (NN - 1);
            float ang = -6.28318530717958647692f * (float)ph / (float)NN;
            float s_, c_;
            __sincosf(ang, &s_, &c_);
            float fv = f[k * KS + t];
            re += fv * c_;
            im += fv * s_;
        }
        sum += re * re + im * im;
    }
    __shared__ float red[256];
    red[threadIdx.x] = sum;
    __syncthreads();
    for (int s = 128; s > 0; s >>= 1) {
        if (threadIdx.x < s)
            red[threadIdx.x] = fmaxf(red[threadIdx.x], red[threadIdx.x + s]);
        __syncthreads();
    }
    // all values >= 0 -> uint bit pattern preserves float order
    if (threadIdx.x == 0) atomicMax(lval, __float_as_uint(red[0]));
}

// ---------------------------------------------------------------------------
// Kernel A: v = signal - H @ w, where w = x + mu*(x - px) formed on the fly.
// One block per 16-row tile of v; 4 wave32s each own a 16-col D tile.
// Per filter k: stage an 80x64 window of w into LDS (VALU transform path)
// plus the 16xWIN Toeplitz slice (async copy), then 20 WMMAs into v8f.
__global__ void __launch_bounds__(128)
k_conv_v(const float* __restrict__ x, const float* __restrict__ px,
         const float* __restrict__ sig, const float* __restrict__ at,
         float* __restrict__ v, float mu) {
    extern __shared__ float lds[];
    float* lB = lds;                 // [WIN][LSTRIDE]
    float* lA = lds + WIN * LSTRIDE; // [16][WINP]

    const int i0   = blockIdx.x * 16;
    const int tid  = threadIdx.x;
    const int lane = tid & 31;
    const int half = lane >> 4;      // 0: lanes 0-15, 1: lanes 16-31
    const int m    = lane & 15;
    const int b0   = (tid >> 5) * 16;

    v8f acc = {};
    for (int k = 0; k < KF; ++k) {
        // async-stage the Toeplitz slice (pure copy, 336 x b128)
        for (int idx = tid; idx < (16 * WINP) / 4; idx += 128)
            stage16(at + (size_t)k * 16 * WINP + 4 * idx, lA + 4 * idx);
        // stage w window: w = x + mu*(x - px) (coalesced float4 loads)
        for (int idx = tid; idx < WIN * 16; idx += 128) {
            int u = idx >> 4, b4 = idx & 15;
            int row = k * NN + ((i0 + (NN - 63) + u) & (NN - 1));
            float4 xv = ((const float4*)x)[row * 16 + b4];
            float4 pv = ((const float4*)px)[row * 16 + b4];
            float4 wv;
            wv.x = xv.x + mu * (xv.x - pv.x);
            wv.y = xv.y + mu * (xv.y - pv.y);
            wv.z = xv.z + mu * (xv.z - pv.z);
            wv.w = xv.w + mu * (xv.w - pv.w);
            *(float4*)&lB[u * LSTRIDE + 4 * b4] = wv;
        }
        stage_wait();
        __syncthreads();
        #pragma unroll
        for (int kk = 0; kk < WIN; kk += 4) {
            v2f a = *(const v2f*)&lA[m * WINP + kk + 2 * half];
            v2f b;
            b.x = lB[(kk + 2 * half) * LSTRIDE + b0 + m];
            b.y = lB[(kk + 1 + 2 * half) * LSTRIDE + b0 + m];
            acc = __builtin_amdgcn_wmma_f32_16x16x4_f32(
                false, a, false, b, (short)0, acc, false, false);
        }
        __syncthreads();
    }
    #pragma unroll
    for (int vg = 0; vg < 8; ++vg) {
        int idx = (i0 + vg + 8 * half) * NB + b0 + m;
        v[idx] = sig[idx] - acc[vg];
    }
}

// ---------------------------------------------------------------------------
// Kernel B: c = w + (1/L) * H^T v ; x_new = soft(c, LAM/L).
// One block per 16-row tile of c (each tile lies within one filter block).
// Both staging paths are pure copies -> fully async global->LDS.
__global__ void __launch_bounds__(128)
k_corr_x(const float* __restrict__ x, const float* __restrict__ px,
         const float* __restrict__ v, const float* __restrict__ at2,
         const unsigned int* __restrict__ lval,
         float* __restrict__ xout, float mu) {
    extern __shared__ float lds[];
    float* lB = lds;                 // [WIN][LSTRIDE]
    float* lA = lds + WIN * LSTRIDE; // [16][WINP]

    const int row0 = blockIdx.x * 16;     // 0..16383
    const int k    = row0 >> 11;          // 2048 rows per filter block
    const int n0   = row0 & (NN - 1);
    const int tid  = threadIdx.x;
    const int lane = tid & 31;
    const int half = lane >> 4;
    const int m    = lane & 15;
    const int b0   = (tid >> 5) * 16;

    for (int idx = tid; idx < (16 * WINP) / 4; idx += 128)
        stage16(at2 + (size_t)k * 16 * WINP + 4 * idx, lA + 4 * idx);
    for (int idx = tid; idx < WIN * 16; idx += 128) {
        int u = idx >> 4, b4 = idx & 15;
        int row = (n0 + u) & (NN - 1);
        stage16(v + (size_t)row * NB + 4 * b4, &lB[u * LSTRIDE + 4 * b4]);
    }
    stage_wait();
    __syncthreads();

    float L    = __uint_as_float(*lval) + 1.0f;
    float invL = 1.0f / L;
    float thr  = LAMBDA / L;

    v8f acc = {};
    #pragma unroll
    for (int kk = 0; kk < WIN; kk += 4) {
        v2f a = *(const v2f*)&lA[m * WINP + kk + 2 * half];
        v2f b;
        b.x = lB[(kk + 2 * half) * LSTRIDE + b0 + m];
        b.y = lB[(kk + 1 + 2 * half) * LSTRIDE + b0 + m];
        acc = __builtin_amdgcn_wmma_f32_16x16x4_f32(
            false, a, false, b, (short)0, acc, false, false);
    }

    #pragma unroll
    for (int vg = 0; vg < 8; ++vg) {
        int idx = (row0 + vg + 8 * half) * NB + b0 + m;
        float xi = x[idx], pi = px[idx];
        float w  = xi + mu * (xi - pi);
        float c  = w + invL * acc[vg];
        float mg = fabsf(c) - thr;
        xout[idx] = (mg > 0.f) ? copysignf(mg, c) : 0.f;
    }
}

// ---------------------------------------------------------------------------
extern "C" void kernel_launch(void* const* d_in, const int* in_sizes, int n_in,
                              void* d_out, int out_size, void* d_ws, size_t ws_size,
                              hipStream_t stream) {
    const float* sig  = (const float*)d_in[0];   // [2048, 64] f32
    const float* dict = (const float*)d_in[1];   // [8, 64]    f32
    float* out = (float*)d_out;                  // [16384, 64] f32
    float* ws  = (float*)d_ws;

    float* X0 = ws + OFF_X0;
    float* X1 = ws + OFF_X1;
    float* V  = ws + OFF_V;
    float* AT = ws + OFF_AT;
    float* AT2 = ws + OFF_AT2;
    unsigned int* LV = (unsigned int*)(ws + OFF_L);

    k_init<<<512, 256, 0, stream>>>(X0, X1, LV, XELEMS / 4);
    k_toep<<<42, 256, 0, stream>>>(dict, AT, AT2);
    k_eig<<<NN / 256, 256, 0, stream>>>(dict, LV);

    const size_t ldsBytes = (size_t)(WIN * LSTRIDE + 16 * WINP) * sizeof(float);
    float* xb[2] = {X0, X1};
    double s = 0.0;
    for (int t = 0; t < T_ITERS; ++t) {
        double sn = (1.0 + sqrt(1.0 + 4.0 * s * s)) * 0.5;
        float mu = (float)((s - 1.0) / sn);
        s = sn;
        float* xc = xb[t & 1];
        float* px = xb[(t + 1) & 1];
        float* tgt = (t == T_ITERS - 1) ? out : px;  // last iter -> d_out
        k_conv_v<<<NN / 16, 128, ldsBytes, stream>>>(xc, px, sig, AT, V, mu);
        k_corr_x<<<(KF * NN) / 16, 128, ldsBytes, stream>>>(xc, px, V, AT2, LV, tgt, mu);
    }
}